// SAGEReg_46883863003259
// MI455X (gfx1250) — compile-verified
//
#include <hip/hip_runtime.h>
#include <hip/hip_bf16.h>

#define NNODES 100000
#define NEDGES 1600000
#define DIM    64          // D_IN == HIDDEN == 64
#define NTILES (NNODES / 16)   // 6250 exact 16-node tiles

typedef float v2f __attribute__((ext_vector_type(2)));
typedef float v8f __attribute__((ext_vector_type(8)));

// ---------------------------------------------------------------------------
// Kernel 1: zero msg_sum + deg workspace (must run every launch; ws is not
// re-poisoned between replays).
// ---------------------------------------------------------------------------
__global__ void sage_zero_kernel(float* __restrict__ p, int n) {
    int i = blockIdx.x * blockDim.x + threadIdx.x;
    if (i < n) p[i] = 0.0f;
}

// ---------------------------------------------------------------------------
// Kernel 2: edge scatter.  16 threads per edge, each moving a float4 chunk of
// x[src] into msg[dst] via global_atomic_add_f32.  Entire working set
// (x + msg + edges ~77MB) is L2-resident on MI455X (192MB), so the random
// gather/scatter runs at L2 bandwidth.
// ---------------------------------------------------------------------------
__global__ __launch_bounds__(256) void sage_scatter_kernel(
    const float* __restrict__ x,
    const long long* __restrict__ ei,   // [2, E] int64: row0 = src, row1 = dst
    float* __restrict__ msg,
    float* __restrict__ deg)
{
    int tid = blockIdx.x * blockDim.x + threadIdx.x;   // E*16 threads
    int e   = tid >> 4;
    int c4  = (tid & 15) << 2;                         // column chunk (0..60 step 4)
    if (e >= NEDGES) return;

    int s = (int)ei[e];
    int d = (int)ei[NEDGES + e];

    float4 v = ((const float4*)x)[s * (DIM / 4) + (c4 >> 2)];
    float* m = msg + (size_t)d * DIM + c4;
    atomicAdd(m + 0, v.x);
    atomicAdd(m + 1, v.y);
    atomicAdd(m + 2, v.z);
    atomicAdd(m + 3, v.w);
    if (c4 == 0) atomicAdd(deg + d, 1.0f);
}

// ---------------------------------------------------------------------------
// Kernel 3: fused  relu(agg@Wl^T + bl + x@Wr^T) @ Wh^T + bh  via fp32 WMMA.
// One wave per 16-node tile; 8 waves (tiles) per 256-thread block.
// 128 x v_wmma_f32_16x16x4_f32 per wave.
// ---------------------------------------------------------------------------
__global__ __launch_bounds__(256, 2) void sage_wmma_kernel(
    const float* __restrict__ x,
    const float* __restrict__ msg,
    const float* __restrict__ deg,
    const float* __restrict__ Wl,   // [64,64] row-major: Wl[h][k]
    const float* __restrict__ bl,   // [64]
    const float* __restrict__ Wr,   // [64,64]
    const float* __restrict__ Wh,   // [1,64]
    const float* __restrict__ bh,   // [1]
    float* __restrict__ out)        // [N]
{
    __shared__ float sWl[DIM * DIM];
    __shared__ float sWr[DIM * DIM];
    __shared__ float sWh[DIM];
    __shared__ float sBl[DIM];

    // Cooperative weight staging into LDS (32KB of the 320KB/WGP pool).
    {
        const float4* gWl = (const float4*)Wl;
        const float4* gWr = (const float4*)Wr;
        float4* lWl = (float4*)sWl;
        float4* lWr = (float4*)sWr;
        for (int i = threadIdx.x; i < DIM * DIM / 4; i += 256) {
            lWl[i] = gWl[i];
            lWr[i] = gWr[i];
        }
        if (threadIdx.x < DIM) {
            sWh[threadIdx.x] = Wh[threadIdx.x];
            sBl[threadIdx.x] = bl[threadIdx.x];
        }
    }
    __syncthreads();

    const int lane = threadIdx.x & 31;
    const int wave = threadIdx.x >> 5;
    const int tile = blockIdx.x * 8 + wave;
    if (tile >= NTILES) return;     // wave-uniform: EXEC stays all-ones

    const int col  = lane & 15;     // N-column within tile / M-row for A frags
    const int hi   = lane >> 4;     // half-wave select (K split for A/B frags)
    const int node = tile * 16 + col;

    // Fuse mean-normalization into the A-fragment load.
    const float rdeg = 1.0f / fmaxf(deg[node], 1.0f);

    // A-fragment layout (16x4 f32): lanes 0-15 -> K={0,1}, lanes 16-31 -> K={2,3}.
    // K-step k covers global K columns [4k, 4k+3]; this lane holds 4k+2*hi, +1.
    v2f aX[16], aA[16];
    const v2f* xRow = (const v2f*)(x   + (size_t)node * DIM);
    const v2f* mRow = (const v2f*)(msg + (size_t)node * DIM);
#pragma unroll
    for (int k = 0; k < 16; ++k) {
        int idx = 2 * k + hi;       // (4k + 2*hi)/2 in float2 units
        aX[k] = xRow[idx];
        aA[k] = mRow[idx] * rdeg;
    }

    // Accumulators: 4 N-tiles of the 16x64 H tile, seeded with bl (broadcast
    // down rows), so both GEMMs + bias accumulate inside WMMA.
    v8f acc[4];
#pragma unroll
    for (int nt = 0; nt < 4; ++nt) {
        float b = sBl[nt * 16 + col];
        acc[nt] = (v8f){b, b, b, b, b, b, b, b};
    }

    // H += agg @ Wl^T   (B frag: B[k][n] = Wl[n][k], n = nt*16+col, k = 4k+2*hi)
#pragma unroll
    for (int k = 0; k < 16; ++k) {
        int kk = 4 * k + 2 * hi;
#pragma unroll
        for (int nt = 0; nt < 4; ++nt) {
            v2f b = *(const v2f*)(sWl + (nt * 16 + col) * DIM + kk);
            acc[nt] = __builtin_amdgcn_wmma_f32_16x16x4_f32(
                false, aA[k], false, b, (short)0, acc[nt], false, false);
        }
    }
    // H += x @ Wr^T
#pragma unroll
    for (int k = 0; k < 16; ++k) {
        int kk = 4 * k + 2 * hi;
#pragma unroll
        for (int nt = 0; nt < 4; ++nt) {
            v2f b = *(const v2f*)(sWr + (nt * 16 + col) * DIM + kk);
            acc[nt] = __builtin_amdgcn_wmma_f32_16x16x4_f32(
                false, aX[k], false, b, (short)0, acc[nt], false, false);
        }
    }

    // Head: out[m] = sum_n relu(H[m][n]) * Wh[n] + bh.
    // C/D layout: VGPR v, lanes 0-15 -> row M=v, lanes 16-31 -> row M=v+8;
    // this lane's column is nt*16+col.  Reduce across the 16 lanes of a half.
    float wh[4];
#pragma unroll
    for (int nt = 0; nt < 4; ++nt) wh[nt] = sWh[nt * 16 + col];
    const float bh0 = bh[0];

    float r[8];
#pragma unroll
    for (int v = 0; v < 8; ++v) {
        float t = 0.0f;
#pragma unroll
        for (int nt = 0; nt < 4; ++nt)
            t += fmaxf(acc[nt][v], 0.0f) * wh[nt];
#pragma unroll
        for (int m = 1; m < 16; m <<= 1)
            t += __shfl_xor(t, m, 32);
        r[v] = t;
    }
    if (col == 0) {
        int base = tile * 16 + hi * 8;   // lanes 0 / 16 write rows 0-7 / 8-15
#pragma unroll
        for (int v = 0; v < 8; ++v) out[base + v] = r[v] + bh0;
    }
}

// ---------------------------------------------------------------------------
extern "C" void kernel_launch(void* const* d_in, const int* in_sizes, int n_in,
                              void* d_out, int out_size, void* d_ws, size_t ws_size,
                              hipStream_t stream) {
    (void)in_sizes; (void)n_in; (void)out_size; (void)ws_size;

    const float*     x  = (const float*)d_in[0];
    const long long* ei = (const long long*)d_in[1];   // int64 per reference
    const float*     Wl = (const float*)d_in[2];
    const float*     bl = (const float*)d_in[3];
    const float*     Wr = (const float*)d_in[4];
    const float*     Wh = (const float*)d_in[5];
    const float*     bh = (const float*)d_in[6];
    float*           out = (float*)d_out;

    float* msg = (float*)d_ws;                     // [N,64]
    float* deg = msg + (size_t)NNODES * DIM;       // [N]

    const int zn = NNODES * (DIM + 1);             // msg + deg contiguous
    sage_zero_kernel<<<(zn + 255) / 256, 256, 0, stream>>>(msg, zn);

    sage_scatter_kernel<<<(NEDGES * 16) / 256, 256, 0, stream>>>(x, ei, msg, deg);

    sage_wmma_kernel<<<(NTILES + 7) / 8, 256, 0, stream>>>(
        x, msg, deg, Wl, bl, Wr, Wh, bh, out);
}